// GlobalAttnSumPool_21792664060772
// MI455X (gfx1250) — compile-verified
//
#include <hip/hip_runtime.h>
#include <math.h>

// ---------------------------------------------------------------------------
// GlobalAttnSumPool for MI455X (gfx1250, wave32)
//   scores = X @ a            (WMMA f32 16x16x4 GEMV)
//   alpha  = softmax(scores)  (global, two tiny reduction kernels)
//   out    = segment_sum(alpha[:,None] * X, I)   (sorted ids, 1 block/segment)
// Memory-bound: ~2.06 GB traffic -> ~88us floor @ 23.3 TB/s.
// ---------------------------------------------------------------------------

typedef __attribute__((ext_vector_type(2))) float v2f;
typedef __attribute__((ext_vector_type(8))) float v8f;

#define WAVES            8
#define TPW              4                  // 16-node tiles per wave
#define K1_THREADS       (WAVES * 32)
#define TILES_PER_BLOCK  (WAVES * TPW)
#define K3_ITER          8

// ---------------- K1: scores[i] = dot(X[i,:], a) via V_WMMA_F32_16X16X4_F32 -
// A tile = 16 nodes x 4 K-slice of X (documented 16x4 f32 layout:
//   lanes 0-15 rows, VGPR0/1 = K0/K1; lanes 16-31 = K2/K3).
// B = `a` slice broadcast across all 16 columns, same (vgpr,half)->K placement
// as A, so every D column holds the identical dot product.
__global__ __launch_bounds__(K1_THREADS)
void k1_scores(const float* __restrict__ X, const float* __restrict__ a,
               float* __restrict__ scores, float* __restrict__ blockmax,
               int nTiles)
{
    const int lane = threadIdx.x & 31;
    const int wave = threadIdx.x >> 5;
    const int half = lane >> 4;   // 0: lanes 0-15, 1: lanes 16-31
    const int lrow = lane & 15;   // matrix row M
    __shared__ float smax[WAVES];

    float wmax = -INFINITY;
    for (int t = 0; t < TPW; ++t) {
        const int tile = blockIdx.x * TILES_PER_BLOCK + t * WAVES + wave;
        if (tile >= nTiles) break;                 // uniform per wave: EXEC stays full
        const float* __restrict__ Xrow = X + ((size_t)tile * 16 + lrow) * 128;
        v8f c = {};
#pragma unroll
        for (int k0 = 0; k0 < 128; k0 += 4) {
            const int k = k0 + (half << 1);
            v2f av; av.x = Xrow[k]; av.y = Xrow[k + 1];
            v2f bv; bv.x = a[k];    bv.y = a[k + 1];
            // (neg_a, A, neg_b, B, c_mod, C, reuse_a, reuse_b)
            c = __builtin_amdgcn_wmma_f32_16x16x4_f32(
                    false, av, false, bv, (short)0, c, false, false);
        }
        // D layout: lane 0 -> rows 0..7 in c[0..7]; lane 16 -> rows 8..15.
        if (lrow == 0) {
            float* sp = scores + (size_t)tile * 16 + half * 8;
#pragma unroll
            for (int j = 0; j < 8; ++j) {
                const float v = c[j];
                sp[j] = v;
                wmax  = fmaxf(wmax, v);
            }
        }
    }
    // wave32 max reduction
#pragma unroll
    for (int off = 16; off > 0; off >>= 1)
        wmax = fmaxf(wmax, __shfl_xor(wmax, off, 32));
    if (lane == 0) smax[wave] = wmax;
    __syncthreads();
    if (threadIdx.x == 0) {
        float b = smax[0];
#pragma unroll
        for (int w = 1; w < WAVES; ++w) b = fmaxf(b, smax[w]);
        blockmax[blockIdx.x] = b;
    }
}

// ---------------- K2: m = max(blockmax) ------------------------------------
__global__ __launch_bounds__(256)
void k2_reduce_max(const float* __restrict__ blockmax, int n,
                   float* __restrict__ mOut)
{
    __shared__ float sm[256];
    float v = -INFINITY;
    for (int i = threadIdx.x; i < n; i += 256) v = fmaxf(v, blockmax[i]);
    sm[threadIdx.x] = v;
    __syncthreads();
    for (int s = 128; s > 0; s >>= 1) {
        if (threadIdx.x < s) sm[threadIdx.x] = fmaxf(sm[threadIdx.x], sm[threadIdx.x + s]);
        __syncthreads();
    }
    if (threadIdx.x == 0) mOut[0] = sm[0];
}

// ---------------- K3: e[i] = exp(s[i]-m) in place; deterministic partials ---
__global__ __launch_bounds__(256)
void k3_exp(float* __restrict__ se, int N, const float* __restrict__ mIn,
            float* __restrict__ bsum)
{
    __shared__ float sm[256];
    const float m = mIn[0];
    float acc = 0.f;
    const int base = blockIdx.x * (256 * K3_ITER) + threadIdx.x;
#pragma unroll
    for (int j = 0; j < K3_ITER; ++j) {
        const int i = base + j * 256;
        if (i < N) {
            const float e = expf(se[i] - m);
            se[i] = e;
            acc += e;
        }
    }
    sm[threadIdx.x] = acc;
    __syncthreads();
    for (int s = 128; s > 0; s >>= 1) {
        if (threadIdx.x < s) sm[threadIdx.x] += sm[threadIdx.x + s];
        __syncthreads();
    }
    if (threadIdx.x == 0) bsum[blockIdx.x] = sm[0];
}

// ---------------- K3b: invZ = 1 / sum(bsum) --------------------------------
__global__ __launch_bounds__(256)
void k3b_reduce_sum(const float* __restrict__ bsum, int n,
                    float* __restrict__ invZ)
{
    __shared__ float sm[256];
    float v = 0.f;
    for (int i = threadIdx.x; i < n; i += 256) v += bsum[i];
    sm[threadIdx.x] = v;
    __syncthreads();
    for (int s = 128; s > 0; s >>= 1) {
        if (threadIdx.x < s) sm[threadIdx.x] += sm[threadIdx.x + s];
        __syncthreads();
    }
    if (threadIdx.x == 0) invZ[0] = 1.0f / sm[0];
}

// ---------------- K4: out[seg,f] = invZ * sum_{i in seg} e[i]*X[i,f] -------
// One block per segment; I is sorted so the node range comes from two binary
// searches. Deterministic (no atomics), writes every output element once.
__global__ __launch_bounds__(128)
void k4_pool(const float* __restrict__ X, const int* __restrict__ I,
             const float* __restrict__ e, const float* __restrict__ invZp,
             float* __restrict__ out, int N)
{
    const int seg = blockIdx.x;
    const int f   = threadIdx.x;  // 0..127
    __shared__ int range[2];
    if (threadIdx.x < 2) {
        const int target = seg + (int)threadIdx.x;   // lower_bound(I, target)
        int lo = 0, hi = N;
        while (lo < hi) {
            const int mid = (lo + hi) >> 1;
            if (I[mid] < target) lo = mid + 1; else hi = mid;
        }
        range[threadIdx.x] = lo;
    }
    __syncthreads();
    const int   start = range[0], end = range[1];
    const float invZ  = invZp[0];

    float acc0 = 0.f, acc1 = 0.f;
    int i = start;
    for (; i + 1 < end; i += 2) {
        if (i + 8 < end)   // gfx1250 global_prefetch_b8
            __builtin_prefetch(&X[(size_t)(i + 8) * 128 + f], 0, 1);
        acc0 = fmaf(e[i],     X[(size_t)i       * 128 + f], acc0);
        acc1 = fmaf(e[i + 1], X[(size_t)(i + 1) * 128 + f], acc1);
    }
    if (i < end)
        acc0 = fmaf(e[i], X[(size_t)i * 128 + f], acc0);
    out[(size_t)seg * 128 + f] = (acc0 + acc1) * invZ;
}

// ---------------------------------------------------------------------------
extern "C" void kernel_launch(void* const* d_in, const int* in_sizes, int n_in,
                              void* d_out, int out_size, void* d_ws, size_t ws_size,
                              hipStream_t stream)
{
    (void)n_in; (void)ws_size;
    const float* X = (const float*)d_in[0];
    const int*   I = (const int*)  d_in[1];
    const float* a = (const float*)d_in[2];

    const int N = in_sizes[1];          // 2,000,000 (multiple of 16)
    const int F = in_sizes[2];          // 128 (hardcoded in kernels)
    const int S = out_size / F;         // 20,000 segments
    float* out = (float*)d_out;

    const int nTiles = N / 16;
    const int nb1 = (nTiles + TILES_PER_BLOCK - 1) / TILES_PER_BLOCK;
    const int nb3 = (N + 256 * K3_ITER - 1) / (256 * K3_ITER);

    // Scratch layout (floats): scores/e [N] | blockmax [nb1] | bsum [nb3] | m | invZ
    float* ws       = (float*)d_ws;
    float* scores   = ws;               // reused as e after K3
    float* blockmax = ws + N;
    float* bsum     = blockmax + nb1;
    float* mPtr     = bsum + nb3;
    float* invZ     = mPtr + 1;

    k1_scores     <<<nb1, K1_THREADS, 0, stream>>>(X, a, scores, blockmax, nTiles);
    k2_reduce_max <<<1,   256,        0, stream>>>(blockmax, nb1, mPtr);
    k3_exp        <<<nb3, 256,        0, stream>>>(scores, N, mPtr, bsum);
    k3b_reduce_sum<<<1,   256,        0, stream>>>(bsum, nb3, invZ);
    k4_pool       <<<S,   128,        0, stream>>>(X, I, scores, invZ, out, N);
}